// WaveEquationG_45578192945904
// MI455X (gfx1250) — compile-verified
//
#include <hip/hip_runtime.h>

// CDNA5 / gfx1250 wave-equation trajectory kernel.
// Roofline: output = 255 frames x 512x512 f32 = 267 MB mandatory writes
// -> ~11.5 us floor at 23.3 TB/s; compute (~0.8 GFLOP) is negligible.
// Frames are chained through d_out so per-step reads (frames k-1,k-2 + coef,
// ~4 MB working set) stay in the 192 MB L2; HBM sees only the writeback.
// Laplacian of each 16x16 tile = T*U + U*T (T = tridiag(1,-2,1)) via 8 chained
// V_WMMA_F32_16X16X4_F32 ops; halo rows/cols are rank-1 VALU fixups.
// T is symmetric, so the vertical-pass A operand equals the horizontal-pass
// B operand per K-chunk: materialize it once per chunk from a constant table.

typedef float v2f __attribute__((ext_vector_type(2)));
typedef float v8f __attribute__((ext_vector_type(8)));

#define NG 512
#define NFRAMES 255

struct T16 {
  float v[16][16];
};
constexpr T16 makeT() {
  T16 t{};
  for (int r = 0; r < 16; ++r)
    for (int c = 0; c < 16; ++c)
      t.v[r][c] = (r == c) ? -2.0f : ((r - c == 1 || c - r == 1) ? 1.0f : 0.0f);
  return t;
}
__constant__ T16 Tmat = makeT();

// coef[i][j] = DT^2 * c(i,j)^2 / DX^2 over the full 512x512 grid
__global__ __launch_bounds__(256) void wave_coef_kernel(const float* __restrict__ alpha,
                                                        float* __restrict__ coef) {
  int idx = blockIdx.x * 256 + threadIdx.x;
  int i = idx >> 9;
  int j = idx & (NG - 1);
  const float step = 10.0f / 511.0f;
  float xi = -5.0f + step * (float)i;
  float yj = -5.0f + step * (float)j;
  float c = 2.0f;
#pragma unroll
  for (int g = 0; g < 4; ++g) {
    float dx = xi - alpha[4 + g];
    float dy = yj - alpha[8 + g];
    c += alpha[g] * expf(-(dx * dx + dy * dy));
  }
  const float dt2_over_dx2 =
      (float)((5e-05 * 5e-05) / ((10.0 / 511.0) * (10.0 / 511.0)));
  coef[idx] = dt2_over_dx2 * c * c;
}

// One leapfrog step: out = 2u - up + coef * Lap(u), zero boundary.
// One wave32 per 16x16 tile; 1024 tiles total.
__global__ __launch_bounds__(256) void wave_step_kernel(const float* __restrict__ u,
                                                        const float* __restrict__ up,
                                                        const float* __restrict__ coef,
                                                        float* __restrict__ out) {
  const int lane = threadIdx.x & 31;
  const int wave = threadIdx.x >> 5;
  const int tile = blockIdx.x * 8 + wave;  // 0..1023
  const int ti = tile >> 5;
  const int tj = tile & 31;
  const int gi0 = ti << 4;
  const int gj0 = tj << 4;
  const int half = lane >> 4;  // K-half select for A/B, M-half select for C/D
  const int n = lane & 15;     // N (column) for B/C operands; M (row) for A

  v8f acc = {0.f, 0.f, 0.f, 0.f, 0.f, 0.f, 0.f, 0.f};

  // 8 chained WMMAs: per K-chunk, one T*U (vertical stencil) and one U*T
  // (horizontal stencil). Operand layouts (V_WMMA_F32_16X16X4_F32, wave32):
  //   A 16x4 : lane&15 = M, element v <-> K = 4ch + 2*half + v
  //   B 4x16 : lane&15 = N, element v <-> K = 4ch + 2*half + v
  //   C/D    : lane&15 = N, element v <-> M = v + 8*half
  // By symmetry of T, the same v2f t serves as A (pass 1) and B (pass 2).
#pragma unroll
  for (int ch = 0; ch < 4; ++ch) {
    const int k0 = 4 * ch + 2 * half;
    v2f t = *(const v2f*)&Tmat.v[n][k0];  // T[n][k0], T[n][k0+1]
    v2f b;
    b.x = u[(size_t)(gi0 + k0) * NG + gj0 + n];      // U[K=k0][N=n]
    b.y = u[(size_t)(gi0 + k0 + 1) * NG + gj0 + n];  // U[K=k0+1][N=n]
    acc = __builtin_amdgcn_wmma_f32_16x16x4_f32(false, t, false, b, (short)0,
                                                acc, false, false);
    // A = two contiguous columns of U at row n -> aligned 8B load
    v2f a = *(const v2f*)&u[(size_t)(gi0 + n) * NG + gj0 + k0];
    acc = __builtin_amdgcn_wmma_f32_16x16x4_f32(false, a, false, t, (short)0,
                                                acc, false, false);
  }

  // ---- halo rank-1 corrections (clamped reads; boundary rows masked below) --
  // Row above tile adds into M=0 (acc[0], lanes 0-15);
  // row below adds into M=15 (acc[7], lanes 16-31).
  {
    int hi = half ? min(gi0 + 16, NG - 1) : max(gi0 - 1, 0);
    float hv = u[(size_t)hi * NG + gj0 + n];
    if (half == 0) acc[0] += hv; else acc[7] += hv;
  }
  // Column left of tile adds into N=0 (lanes 0 and 16); right into N=15.
  if (n == 0) {
    int cj = max(gj0 - 1, 0);
#pragma unroll
    for (int v = 0; v < 8; ++v)
      acc[v] += u[(size_t)(gi0 + 8 * half + v) * NG + cj];
  }
  if (n == 15) {
    int cj = min(gj0 + 16, NG - 1);
#pragma unroll
    for (int v = 0; v < 8; ++v)
      acc[v] += u[(size_t)(gi0 + 8 * half + v) * NG + cj];
  }

  // ---- leapfrog combine + boundary mask; C/D layout: M = v+8*half, N = n ----
#pragma unroll
  for (int v = 0; v < 8; ++v) {
    int gi = gi0 + 8 * half + v;
    int gj = gj0 + n;
    size_t idx = (size_t)gi * NG + gj;
    float val = 2.0f * u[idx] - up[idx] + coef[idx] * acc[v];
    bool bdry = (gi == 0) | (gi == NG - 1) | (gj == 0) | (gj == NG - 1);
    out[idx] = bdry ? 0.0f : val;
  }
}

extern "C" void kernel_launch(void* const* d_in, const int* in_sizes, int n_in,
                              void* d_out, int out_size, void* d_ws, size_t ws_size,
                              hipStream_t stream) {
  (void)in_sizes; (void)n_in; (void)out_size; (void)ws_size;
  const float* u0 = (const float*)d_in[0];
  const float* alpha = (const float*)d_in[1];
  float* out = (float*)d_out;
  float* coef = (float*)d_ws;  // 512*512 floats = 1 MiB scratch

  wave_coef_kernel<<<(NG * NG) / 256, 256, 0, stream>>>(alpha, coef);

  const size_t F = (size_t)NG * NG;
  // Frame k-1 (state after k steps) lives at out + (k-1)*F.
  // Step 1: u = u0, u_prev = u0 -> frame 0
  wave_step_kernel<<<128, 256, 0, stream>>>(u0, u0, coef, out);
  // Step 2: u = frame0, u_prev = u0 -> frame 1
  wave_step_kernel<<<128, 256, 0, stream>>>(out, u0, coef, out + F);
  // Steps 3..255: read back from d_out (L2-resident), write next frame.
  for (int k = 3; k <= NFRAMES; ++k)
    wave_step_kernel<<<128, 256, 0, stream>>>(out + (size_t)(k - 2) * F,
                                              out + (size_t)(k - 3) * F,
                                              coef,
                                              out + (size_t)(k - 1) * F);
}